// VAE_61538291417372
// MI455X (gfx1250) — compile-verified
//
#include <hip/hip_runtime.h>
#include <hip/hip_bf16.h>
#include <math.h>

// ---------------------------------------------------------------------------
// CDNA5 (gfx1250) fp32 WMMA GRU-VAE, round 4.
//   V_WMMA_F32_16X16X4_F32:  D(16x16 f32, 8 VGPR) = A(16x4) * B(4x16) + C
//   A frag:  lane l -> row = l&15,  k = kbase + 2*(l>>4) + {0,1}
//   B frag:  lane l -> col = l&15,  k = kbase + 2*(l>>4) + {0,1}
//   C frag:  lane l, reg i -> (M = i + 8*(l>>4), N = l&15)
// Weights are re-packed once into per-fragment contiguous order; hot loops
// use per-wave base pointers + constant offsets (immediate-offset b64 loads).
// All tile staging and result stores are contiguous float4 (b128) transfers
// through LDS.
// ---------------------------------------------------------------------------

typedef __attribute__((ext_vector_type(2))) float v2f;
typedef __attribute__((ext_vector_type(8))) float v8f;

#define BATCH 2048
#define SEQ_LEN 100
#define SEQ_DIM 32
#define INPUT_SIZE 128
#define H 256
#define ZDIM 64

__device__ __forceinline__ v8f wmma4(v2f a, v2f b, v8f c) {
  return __builtin_amdgcn_wmma_f32_16x16x4_f32(
      /*neg_a=*/false, a, /*neg_b=*/false, b,
      /*c_mod=*/(short)0, c, /*reuse_a=*/false, /*reuse_b=*/false);
}

// Generic (cold-path) fragment loaders.
__device__ __forceinline__ v2f load_a(const float* __restrict__ src, int stride,
                                      int k, int lane) {
  const int row = lane & 15;
  const int kk  = k + ((lane >> 4) << 1);
  v2f a;
  a.x = src[row * stride + kk];
  a.y = src[row * stride + kk + 1];
  return a;
}
__device__ __forceinline__ v2f load_bp(const float* __restrict__ P, int kfrags,
                                       int ntile, int kstep, int lane) {
  return *(const v2f*)(P + (((ntile * kfrags) + kstep) << 6) + (lane << 1));
}

// Hardware-trans activations: v_exp_f32 + v_rcp_f32.
__device__ __forceinline__ float sigm(float x) {
  return __builtin_amdgcn_rcpf(1.0f + __expf(-x));
}
__device__ __forceinline__ float tanh_fast(float x) {
  return 1.0f - 2.0f * __builtin_amdgcn_rcpf(1.0f + __expf(2.0f * x));
}

__global__ void zero_kernel(float* __restrict__ p, int n) {
  int i = blockIdx.x * blockDim.x + threadIdx.x;
  if (i < n) p[i] = 0.0f;
}

// Re-pack a row-major (K x N) weight into WMMA B-fragment order.
__global__ void pack_kernel(const float* __restrict__ W, int K, int N,
                            float* __restrict__ P) {
  const int total  = K * N;
  const int kfrags = K >> 2;
  for (int i = blockIdx.x * blockDim.x + threadIdx.x; i < total;
       i += gridDim.x * blockDim.x) {
    const int frag  = i & 63;
    const int rest  = i >> 6;
    const int kstep = rest % kfrags;
    const int ntile = rest / kfrags;
    const int l     = frag >> 1;
    const int half  = frag & 1;
    const int col   = (ntile << 4) + (l & 15);
    const int kk    = (kstep << 2) + ((l >> 4) << 1) + half;
    P[i] = W[kk * N + col];
  }
}

// ---------------------------------------------------------------------------
// One GRU step for a 32-row batch tile x 128-col half of H.
// Block = 256 threads = 8 waves; wave w owns cols [16w,16w+16) of this half,
// for BOTH 16-row M tiles (B fragment reused across the two WMMAs).
// grid = (BATCH/32, 2).
// If pW_emb != nullptr: xt = relu(xin(32 x 32) @ W_emb(32 x 128) + b_emb)
// else:                 xt = xin tile (32 x 128), rows contiguous (stride 128).
// ---------------------------------------------------------------------------
__global__ __launch_bounds__(256)
void gru_step_kernel(const float* __restrict__ xin, int xin_stride,
                     const float* __restrict__ pW_emb, const float* __restrict__ b_emb,
                     const float* __restrict__ pWih, const float* __restrict__ pWhh,
                     const float* __restrict__ bih, const float* __restrict__ bhh,
                     const float* __restrict__ h_in, float* __restrict__ h_out) {
  __shared__ float sX[32 * 128];   // input tile (post-embed)
  __shared__ float sH[32 * 256];   // previous hidden tile; reused for h_out
  __shared__ float sXR[32 * 32];   // raw x tile (embed path)
  const int tid   = threadIdx.x;
  const int lane  = tid & 31;
  const int wave  = tid >> 5;          // 0..7
  const int m0    = blockIdx.x * 32;
  const int nbase = blockIdx.y * 128;  // 0 or 128

  // Stage h tile: 32 contiguous rows -> contiguous 8192 floats, b128 copies.
  {
    const float4* src = (const float4*)(h_in + (size_t)m0 * H);
    float4* dst = (float4*)sH;
    for (int i = tid; i < (32 * 256) / 4; i += 256) dst[i] = src[i];
  }
  if (pW_emb != nullptr) {
    // Stage raw x tile 32x32: one b128 per thread (rows stride xin_stride).
    const int r = tid >> 3, c = tid & 7;
    *((float4*)sXR + tid) =
        *(const float4*)(xin + (size_t)(m0 + r) * xin_stride + c * 4);
  } else {
    // Decoder input: 32 contiguous rows of 128 -> contiguous 4096 floats.
    const float4* src = (const float4*)(xin + (size_t)m0 * INPUT_SIZE);
    float4* dst = (float4*)sX;
    for (int i = tid; i < (32 * 128) / 4; i += 256) dst[i] = src[i];
  }
  __syncthreads();

  if (pW_emb != nullptr) {
    // 16 embed tiles (2 M x 8 N); wave handles tiles wave and wave+8.
    for (int tt = wave; tt < 16; tt += 8) {
      const int mt = tt >> 3;
      const int n0 = (tt & 7) * 16;
      const float* axr = sXR + (mt * 16 + (lane & 15)) * 32 + ((lane >> 4) << 1);
      const float* bbp = pW_emb + ((tt & 7) * 8) * 64 + (lane << 1);
      v8f acc = {};
#pragma unroll
      for (int ks = 0; ks < 8; ks++)
        acc = wmma4(*(const v2f*)(axr + ks * 4), *(const v2f*)(bbp + ks * 64), acc);
      const int col  = n0 + (lane & 15);
      const float bb = b_emb[col];
#pragma unroll
      for (int i = 0; i < 8; i++) {
        const int row = i + ((lane >> 4) << 3);
        const float v = acc[i] + bb;
        sX[(mt * 16 + row) * 128 + col] = v > 0.0f ? v : 0.0f;
      }
    }
    __syncthreads();
  }

  const int n0    = wave * 16;          // col within this half
  const int ntile = (nbase + n0) >> 4;  // 0..15 gate-r tile index
  // Per-wave base pointers: all hot-loop loads below use constant offsets.
  const int arow = (lane & 15);
  const int akof = ((lane >> 4) << 1);
  const float* axb = sX + arow * 128 + akof;            // A from input tile
  const float* ahb = sH + arow * 256 + akof;            // A from hidden tile
  const float* bri = pWih + ((ntile)*32) * 64 + (lane << 1);
  const float* bzi = pWih + ((ntile + 16) * 32) * 64 + (lane << 1);
  const float* bni = pWih + ((ntile + 32) * 32) * 64 + (lane << 1);
  const float* brh = pWhh + ((ntile)*64) * 64 + (lane << 1);
  const float* bzh = pWhh + ((ntile + 16) * 64) * 64 + (lane << 1);
  const float* bnh = pWhh + ((ntile + 32) * 64) * 64 + (lane << 1);

  v8f ar0 = {}, ar1 = {}, az0 = {}, az1 = {}, ai0 = {}, ai1 = {}, ah0 = {}, ah1 = {};

  // Input-side gates, K = 128 (32 k-fragments).
#pragma unroll 8
  for (int ks = 0; ks < 32; ks++) {
    const v2f a0 = *(const v2f*)(axb + ks * 4);
    const v2f a1 = *(const v2f*)(axb + 16 * 128 + ks * 4);
    const v2f br = *(const v2f*)(bri + ks * 64);
    const v2f bz = *(const v2f*)(bzi + ks * 64);
    const v2f bi = *(const v2f*)(bni + ks * 64);
    ar0 = wmma4(a0, br, ar0); ar1 = wmma4(a1, br, ar1);
    az0 = wmma4(a0, bz, az0); az1 = wmma4(a1, bz, az1);
    ai0 = wmma4(a0, bi, ai0); ai1 = wmma4(a1, bi, ai1);
  }
  // Hidden-side gates, K = 256 (64 k-fragments).
#pragma unroll 8
  for (int ks = 0; ks < 64; ks++) {
    const v2f a0 = *(const v2f*)(ahb + ks * 4);
    const v2f a1 = *(const v2f*)(ahb + 16 * 256 + ks * 4);
    const v2f br = *(const v2f*)(brh + ks * 64);
    const v2f bz = *(const v2f*)(bzh + ks * 64);
    const v2f bh = *(const v2f*)(bnh + ks * 64);
    ar0 = wmma4(a0, br, ar0); ar1 = wmma4(a1, br, ar1);
    az0 = wmma4(a0, bz, az0); az1 = wmma4(a1, bz, az1);
    ah0 = wmma4(a0, bh, ah0); ah1 = wmma4(a1, bh, ah1);
  }

  // All A-reads of sH are done; combine results in place (own cols only).
  __syncthreads();

  const int col  = nbase + n0 + (lane & 15);   // global col in [0,256)
  const float br = bih[col] + bhh[col];
  const float bz = bih[H + col] + bhh[H + col];
  const float bi = bih[2 * H + col];
  const float bh = bhh[2 * H + col];
#pragma unroll
  for (int i = 0; i < 8; i++) {
    const int row = i + ((lane >> 4) << 3);
    {  // M tile 0
      const float r  = sigm(ar0[i] + br);
      const float u  = sigm(az0[i] + bz);
      const float n  = tanh_fast(ai0[i] + bi + r * (ah0[i] + bh));
      const float ho = sH[row * 256 + col];
      sH[row * 256 + col] = (1.0f - u) * n + u * ho;
    }
    {  // M tile 1
      const float r  = sigm(ar1[i] + br);
      const float u  = sigm(az1[i] + bz);
      const float n  = tanh_fast(ai1[i] + bi + r * (ah1[i] + bh));
      const float ho = sH[(16 + row) * 256 + col];
      sH[(16 + row) * 256 + col] = (1.0f - u) * n + u * ho;
    }
  }
  __syncthreads();

  // Stream out the owned 32 x 128 half as b128 stores.
  for (int i = tid; i < 32 * 32; i += 256) {
    const int row = i >> 5, c4 = (i & 31) * 4;
    *(float4*)(h_out + (size_t)(m0 + row) * H + nbase + c4) =
        *(const float4*)(sH + row * 256 + nbase + c4);
  }
}

// ---------------------------------------------------------------------------
// Encoder head: stats -> (mu, logvar, z) -> h_dec0 -> hid_inp -> inp0.
// Block = 128 threads = 4 waves, one 16-row batch tile.  Runs once.
// ---------------------------------------------------------------------------
__global__ __launch_bounds__(128)
void enc_head_kernel(const float* __restrict__ h_fwd, const float* __restrict__ h_bwd,
                     const float* __restrict__ pW_enc, const float* __restrict__ b_enc,
                     const float* __restrict__ eps,
                     const float* __restrict__ pW_z2h, const float* __restrict__ b_z2h,
                     const float* __restrict__ pW_h2i, const float* __restrict__ b_h2i,
                     const float* __restrict__ b_ed,
                     float* __restrict__ mu_out, float* __restrict__ lv_out,
                     float* __restrict__ h_dec0_out, float* __restrict__ inp0) {
  __shared__ float sF[16 * 256];
  __shared__ float sB[16 * 256];
  __shared__ float sS[16 * 128];   // stats
  __shared__ float sZ[16 * 64];    // z
  __shared__ float sD[16 * 256];   // h_dec0
  const int tid  = threadIdx.x;
  const int lane = tid & 31;
  const int wave = tid >> 5;       // 0..3
  const int m0   = blockIdx.x * 16;

  {
    const float4* f = (const float4*)(h_fwd + (size_t)m0 * H);
    const float4* b = (const float4*)(h_bwd + (size_t)m0 * H);
    for (int i = tid; i < (16 * 256) / 4; i += 128) {
      ((float4*)sF)[i] = f[i];
      ((float4*)sB)[i] = b[i];
    }
  }
  __syncthreads();

  // stats = [h_fwd | h_bwd] @ W_enc(512 x 128) + b_enc.  kfrags = 128.
  for (int tt = wave; tt < 8; tt += 4) {
    const int n0 = tt * 16;
    v8f acc = {};
#pragma unroll 8
    for (int ks = 0; ks < 64; ks++)
      acc = wmma4(load_a(sF, 256, ks * 4, lane), load_bp(pW_enc, 128, tt, ks, lane), acc);
#pragma unroll 8
    for (int ks = 64; ks < 128; ks++)
      acc = wmma4(load_a(sB, 256, (ks - 64) * 4, lane), load_bp(pW_enc, 128, tt, ks, lane), acc);
    const int col  = n0 + (lane & 15);
    const float bb = b_enc[col];
#pragma unroll
    for (int i = 0; i < 8; i++) {
      const int row = i + ((lane >> 4) << 3);
      sS[row * 128 + col] = acc[i] + bb;
    }
  }
  __syncthreads();

  // Reparameterize: z = mu + exp(0.5*logvar)*eps; emit mu / logvar.
  for (int i = tid; i < 16 * 64; i += 128) {
    const int row = i >> 6, col = i & 63;
    const float mu = sS[row * 128 + col];
    const float lv = sS[row * 128 + 64 + col];
    mu_out[(m0 + row) * ZDIM + col] = mu;
    lv_out[(m0 + row) * ZDIM + col] = lv;
    sZ[row * 64 + col] = mu + __expf(0.5f * lv) * eps[(m0 + row) * ZDIM + col];
  }
  __syncthreads();

  // h_dec0 = relu(z @ W_z2h(64 x 256) + b).  kfrags = 16, 16 col tiles.
  for (int tt = wave; tt < 16; tt += 4) {
    const int n0 = tt * 16;
    v8f acc = {};
#pragma unroll
    for (int ks = 0; ks < 16; ks++)
      acc = wmma4(load_a(sZ, 64, ks * 4, lane), load_bp(pW_z2h, 16, tt, ks, lane), acc);
    const int col  = n0 + (lane & 15);
    const float bb = b_z2h[col];
#pragma unroll
    for (int i = 0; i < 8; i++) {
      const int row = i + ((lane >> 4) << 3);
      float v = acc[i] + bb;
      v = v > 0.0f ? v : 0.0f;
      sD[row * 256 + col] = v;
      h_dec0_out[(m0 + row) * H + col] = v;
    }
  }
  __syncthreads();

  // hid_inp = relu(h_dec0 @ W_h2i(256 x 64) + b) -> inp0 cols [64,128).
  {
    const int n0 = wave * 16;
    v8f acc = {};
#pragma unroll 8
    for (int ks = 0; ks < 64; ks++)
      acc = wmma4(load_a(sD, 256, ks * 4, lane), load_bp(pW_h2i, 64, wave, ks, lane), acc);
    const int col  = n0 + (lane & 15);
    const float bb = b_h2i[col];
#pragma unroll
    for (int i = 0; i < 8; i++) {
      const int row = i + ((lane >> 4) << 3);
      const float v = acc[i] + bb;
      inp0[(m0 + row) * 128 + 64 + col] = v > 0.0f ? v : 0.0f;
    }
  }
  // inp0 cols [0,64) = relu(0 @ W_ed + b_ed) = relu(b_ed): batch-independent.
  for (int i = tid; i < 16 * 64; i += 128) {
    const int row = i >> 6, col = i & 63;
    const float v = b_ed[col];
    inp0[(m0 + row) * 128 + col] = v > 0.0f ? v : 0.0f;
  }
}

// ---------------------------------------------------------------------------
// Decoder tail per step: x_cur = h2@W_dec + b  (-> output),
// inp_next = [ relu(x_cur@W_ed + b) | relu(h2@W_h2i + b) ].
// Block = 128 threads = 4 waves, one 16-row batch tile.
// Results are assembled in LDS and streamed out as b128 stores.
// ---------------------------------------------------------------------------
__global__ __launch_bounds__(128)
void dec_out_kernel(const float* __restrict__ h2,
                    const float* __restrict__ pW_dec, const float* __restrict__ b_dec,
                    const float* __restrict__ pW_ed, const float* __restrict__ b_ed,
                    const float* __restrict__ pW_h2i, const float* __restrict__ b_h2i,
                    float* __restrict__ xout,        // + t*32, row stride 3200
                    float* __restrict__ inp_next) {
  __shared__ float sH[16 * 256];
  __shared__ float sXC[16 * 32];
  __shared__ float sI[16 * 128];   // assembled inp_next tile
  const int tid  = threadIdx.x;
  const int lane = tid & 31;
  const int wave = tid >> 5;   // 0..3
  const int m0   = blockIdx.x * 16;

  {
    const float4* src = (const float4*)(h2 + (size_t)m0 * H);
    for (int i = tid; i < (16 * 256) / 4; i += 128) ((float4*)sH)[i] = src[i];
  }
  __syncthreads();

  const float* ahb = sH + (lane & 15) * 256 + ((lane >> 4) << 1);

  // relu(h2 @ W_h2i) -> sI cols [64,128), one 16-col tile per wave.
  {
    const float* bb64 = pW_h2i + (wave * 64) * 64 + (lane << 1);
    v8f acc = {};
#pragma unroll 8
    for (int ks = 0; ks < 64; ks++)
      acc = wmma4(*(const v2f*)(ahb + ks * 4), *(const v2f*)(bb64 + ks * 64), acc);
    const int col  = wave * 16 + (lane & 15);
    const float bb = b_h2i[col];
#pragma unroll
    for (int i = 0; i < 8; i++) {
      const int row = i + ((lane >> 4) << 3);
      const float v = acc[i] + bb;
      sI[row * 128 + 64 + col] = v > 0.0f ? v : 0.0f;
    }
  }
  // x_cur = h2 @ W_dec + b_dec (32 cols): waves 0,1.
  if (wave < 2) {
    const float* bb64 = pW_dec + (wave * 64) * 64 + (lane << 1);
    v8f acc = {};
#pragma unroll 8
    for (int ks = 0; ks < 64; ks++)
      acc = wmma4(*(const v2f*)(ahb + ks * 4), *(const v2f*)(bb64 + ks * 64), acc);
    const int col  = wave * 16 + (lane & 15);
    const float bb = b_dec[col];
#pragma unroll
    for (int i = 0; i < 8; i++) {
      const int row = i + ((lane >> 4) << 3);
      sXC[row * 32 + col] = acc[i] + bb;
    }
  }
  __syncthreads();

  // Stream x_cur out (16 rows x 32 cols, b128): one float4 per thread.
  {
    const int row = tid >> 3, c4 = (tid & 7) * 4;
    *(float4*)(xout + (size_t)(m0 + row) * (SEQ_LEN * SEQ_DIM) + c4) =
        *(const float4*)(sXC + row * 32 + c4);
  }
  // relu(x_cur @ W_ed) -> sI cols [0,64), one tile per wave, kfrags = 8.
  {
    const float* axc  = sXC + (lane & 15) * 32 + ((lane >> 4) << 1);
    const float* bb64 = pW_ed + (wave * 8) * 64 + (lane << 1);
    v8f acc = {};
#pragma unroll
    for (int ks = 0; ks < 8; ks++)
      acc = wmma4(*(const v2f*)(axc + ks * 4), *(const v2f*)(bb64 + ks * 64), acc);
    const int col  = wave * 16 + (lane & 15);
    const float bb = b_ed[col];
#pragma unroll
    for (int i = 0; i < 8; i++) {
      const int row = i + ((lane >> 4) << 3);
      const float v = acc[i] + bb;
      sI[row * 128 + col] = v > 0.0f ? v : 0.0f;
    }
  }
  __syncthreads();

  // Stream inp_next out: contiguous 2048 floats, b128.
  {
    float4* dst = (float4*)(inp_next + (size_t)m0 * INPUT_SIZE);
    for (int i = tid; i < (16 * 128) / 4; i += 128)
      dst[i] = ((const float4*)sI)[i];
  }
}

// ---------------------------------------------------------------------------
extern "C" void kernel_launch(void* const* d_in, const int* in_sizes, int n_in,
                              void* d_out, int out_size, void* d_ws, size_t ws_size,
                              hipStream_t stream) {
  const float* x     = (const float*)d_in[0];
  const float* eps   = (const float*)d_in[1];
  const float* W_ee  = (const float*)d_in[2];
  const float* b_ee  = (const float*)d_in[3];
  const float* Wih_f = (const float*)d_in[4];
  const float* Whh_f = (const float*)d_in[5];
  const float* bih_f = (const float*)d_in[6];
  const float* bhh_f = (const float*)d_in[7];
  const float* Wih_b = (const float*)d_in[8];
  const float* Whh_b = (const float*)d_in[9];
  const float* bih_b = (const float*)d_in[10];
  const float* bhh_b = (const float*)d_in[11];
  const float* W_enc = (const float*)d_in[12];
  const float* b_enc = (const float*)d_in[13];
  const float* W_z2h = (const float*)d_in[14];
  const float* b_z2h = (const float*)d_in[15];
  const float* W_h2i = (const float*)d_in[16];
  const float* b_h2i = (const float*)d_in[17];
  const float* Wih_d = (const float*)d_in[18];
  const float* Whh_d = (const float*)d_in[19];
  const float* bih_d = (const float*)d_in[20];
  const float* bhh_d = (const float*)d_in[21];
  const float* W_ed  = (const float*)d_in[22];
  const float* b_ed  = (const float*)d_in[23];
  const float* W_dec = (const float*)d_in[24];
  const float* b_dec = (const float*)d_in[25];

  float* out     = (float*)d_out;
  float* x_recon = out;                                      // 2048*100*32
  float* mu_out  = out + (size_t)BATCH * SEQ_LEN * SEQ_DIM;  // 2048*64
  float* lv_out  = mu_out + (size_t)BATCH * ZDIM;

  const int HB = BATCH * H;           // 524288 floats per hidden buffer
  const int IB = BATCH * INPUT_SIZE;  // 262144
  float* hZ   = (float*)d_ws;
  float* hA   = hZ + HB;
  float* hB   = hA + HB;
  float* hBwd = hB + HB;
  float* inpA = hBwd + HB;
  float* inpB = inpA + IB;
  // Packed weight region (sizes == K*N of each matrix).
  float* pWee   = inpB + IB;
  float* pWih_f = pWee + 32 * 128;
  float* pWhh_f = pWih_f + 128 * 768;
  float* pWih_b = pWhh_f + 256 * 768;
  float* pWhh_b = pWih_b + 128 * 768;
  float* pWenc  = pWhh_b + 256 * 768;
  float* pWz2h  = pWenc + 512 * 128;
  float* pWh2i  = pWz2h + 64 * 256;
  float* pWih_d = pWh2i + 256 * 64;
  float* pWhh_d = pWih_d + 128 * 768;
  float* pWed   = pWhh_d + 256 * 768;
  float* pWdec  = pWed + 32 * 64;

  zero_kernel<<<(HB + 255) / 256, 256, 0, stream>>>(hZ, HB);

  // One-time weight re-pack into WMMA fragment order (L2-resident afterwards).
  pack_kernel<<<64, 256, 0, stream>>>(W_ee,  32, 128, pWee);
  pack_kernel<<<256, 256, 0, stream>>>(Wih_f, 128, 768, pWih_f);
  pack_kernel<<<256, 256, 0, stream>>>(Whh_f, 256, 768, pWhh_f);
  pack_kernel<<<256, 256, 0, stream>>>(Wih_b, 128, 768, pWih_b);
  pack_kernel<<<256, 256, 0, stream>>>(Whh_b, 256, 768, pWhh_b);
  pack_kernel<<<128, 256, 0, stream>>>(W_enc, 512, 128, pWenc);
  pack_kernel<<<64, 256, 0, stream>>>(W_z2h, 64, 256, pWz2h);
  pack_kernel<<<64, 256, 0, stream>>>(W_h2i, 256, 64, pWh2i);
  pack_kernel<<<256, 256, 0, stream>>>(Wih_d, 128, 768, pWih_d);
  pack_kernel<<<256, 256, 0, stream>>>(Whh_d, 256, 768, pWhh_d);
  pack_kernel<<<16, 256, 0, stream>>>(W_ed,  32, 64,  pWed);
  pack_kernel<<<32, 256, 0, stream>>>(W_dec, 256, 32, pWdec);

  const dim3 gGru(BATCH / 32, 2), bGru(256);
  const dim3 gSm(BATCH / 16), bSm(128);

  // ----- encoder forward scan -----
  const float* hin = hZ;
  float* hout = hA;
  for (int t = 0; t < SEQ_LEN; ++t) {
    gru_step_kernel<<<gGru, bGru, 0, stream>>>(
        x + (size_t)t * SEQ_DIM, SEQ_LEN * SEQ_DIM, pWee, b_ee,
        pWih_f, pWhh_f, bih_f, bhh_f, hin, hout);
    hin = hout;
    hout = (hout == hA) ? hB : hA;
  }
  // hin now holds the final forward hidden state.

  // ----- encoder backward cell (single step on xe[:, -1]) -----
  gru_step_kernel<<<gGru, bGru, 0, stream>>>(
      x + (size_t)(SEQ_LEN - 1) * SEQ_DIM, SEQ_LEN * SEQ_DIM, pWee, b_ee,
      pWih_b, pWhh_b, bih_b, bhh_b, hZ, hBwd);

  // ----- encoder head -----
  float* hDec = (hin == hA) ? hB : hA;   // free buffer
  enc_head_kernel<<<gSm, bSm, 0, stream>>>(
      hin, hBwd, pWenc, b_enc, eps, pWz2h, b_z2h, pWh2i, b_h2i, b_ed,
      mu_out, lv_out, hDec, inpA);

  // ----- decoder scan -----
  const float* hdin = hDec;
  float* hdout = (hDec == hA) ? hB : hA;
  const float* ip = inpA;
  float* ipn = inpB;
  for (int t = 0; t < SEQ_LEN; ++t) {
    gru_step_kernel<<<gGru, bGru, 0, stream>>>(
        ip, INPUT_SIZE, nullptr, nullptr,
        pWih_d, pWhh_d, bih_d, bhh_d, hdin, hdout);
    dec_out_kernel<<<gSm, bSm, 0, stream>>>(
        hdout, pWdec, b_dec, pWed, b_ed, pWh2i, b_h2i,
        x_recon + (size_t)t * SEQ_DIM, ipn);
    const float* th = hdin; hdin = hdout; hdout = (float*)th;
    const float* ti = ip;   ip = ipn;     ipn = (float*)ti;
  }
}